// RungeKuttaIntegration_40492951667500
// MI455X (gfx1250) — compile-verified
//
#include <hip/hip_runtime.h>

// RK4 motion-field integration, MI455X (gfx1250, wave32).
//
// Strategy: the 66.4MB motion field is L2-resident (192MB L2), so the kernel
// is bound by L2 gather rate + latency hiding. Pre-pass interleaves the two
// channel planes into (H,W,2) so each RK4 sample is a single global_load_b64
// gather (halves VMEM instruction count vs planar). The pre-pass stages data
// through LDS with the CDNA5 async global->LDS path (ASYNCcnt) when the
// toolchain exposes it. Output is streamed with non-temporal stores to avoid
// evicting the motion field from L2.

#if defined(__HIP_DEVICE_COMPILE__)
#if __has_builtin(__builtin_amdgcn_global_load_async_to_lds_b32)
#define USE_ASYNC_LDS 1
#endif
#endif
#ifndef USE_ASYNC_LDS
#define USE_ASYNC_LDS 0
#endif

namespace {

constexpr int kH  = 2160;
constexpr int kW  = 3840;
constexpr int kHW = kH * kW;

#if USE_ASYNC_LDS
typedef __attribute__((address_space(1))) int gas_int;  // global (AS1)
typedef __attribute__((address_space(3))) int las_int;  // LDS    (AS3)
#endif

__device__ __forceinline__ int clamp_round(float c, float maxv) {
  // jnp.round == round-half-to-even == v_rndne_f32 (rintf), then clip.
  float f = fminf(fmaxf(rintf(c), 0.0f), maxv);
  return (int)f;
}

__device__ __forceinline__ float2 samp_interleaved(const float2* __restrict__ mi,
                                                   float cx, float cy) {
  const int ix = clamp_round(cx, (float)(kW - 1));
  const int iy = clamp_round(cy, (float)(kH - 1));
  return mi[iy * kW + ix];                       // one global_load_b64 gather
}

__device__ __forceinline__ float2 samp_planar(const float* __restrict__ m,
                                              float cx, float cy) {
  const int ix  = clamp_round(cx, (float)(kW - 1));
  const int iy  = clamp_round(cy, (float)(kH - 1));
  const int idx = iy * kW + ix;
  return make_float2(m[idx], m[kHW + idx]);      // two global_load_b32 gathers
}

}  // namespace

// ---- Pre-pass: (2,H,W) planar -> (H,W,2) interleaved -----------------------
// Async variant: each lane issues two GLOBAL_LOAD_ASYNC_TO_LDS_B32 (one per
// channel) into wave-private LDS slots, drains ASYNCcnt, then emits the
// interleaved b64 store. No barrier needed: every lane reads back only the
// LDS bytes its own wave's async ops wrote.
__global__ __launch_bounds__(256) void rk4_interleave_kernel(
    const float* __restrict__ motion, float2* __restrict__ mi) {
  const int tid = threadIdx.x;
  const int idx = blockIdx.x * 256 + tid;
  if (idx >= kHW) return;
#if USE_ASYNC_LDS
  __shared__ float stage[512];  // [0,256): channel x, [256,512): channel y
  __builtin_amdgcn_global_load_async_to_lds_b32(
      (gas_int*)(motion + idx),
      (las_int*)&stage[tid],
      /*imm_offset=*/0, /*cpol=*/0);
  __builtin_amdgcn_global_load_async_to_lds_b32(
      (gas_int*)(motion + kHW + idx),
      (las_int*)&stage[256 + tid],
      /*imm_offset=*/0, /*cpol=*/0);
#if __has_builtin(__builtin_amdgcn_s_wait_asynccnt)
  __builtin_amdgcn_s_wait_asynccnt(0);
#else
  asm volatile("s_wait_asynccnt 0x0" ::: "memory");
#endif
  asm volatile("" ::: "memory");  // keep the LDS read-back below the wait
  mi[idx] = make_float2(stage[tid], stage[256 + tid]);
#else
  // Fallback: one-shot streaming read (NT); the interleaved copy is the hot
  // data and is stored with default (RT) policy so it stays resident in L2.
  const float mx = __builtin_nontemporal_load(&motion[idx]);
  const float my = __builtin_nontemporal_load(&motion[kHW + idx]);
  mi[idx] = make_float2(mx, my);
#endif
}

// ---- Main RK4 integration --------------------------------------------------
template <bool INTERLEAVED>
__global__ __launch_bounds__(256) void rk4_integrate_kernel(
    const void* __restrict__ field,
    const int*  __restrict__ n_steps_ptr,
    float*      __restrict__ out) {
  const int x   = blockIdx.x * 256 + threadIdx.x;
  const int y   = blockIdx.y;
  const int idx = y * kW + x;

  const float2* __restrict__ mi = (const float2*)field;
  const float*  __restrict__ mp = (const float*)field;

  if (INTERLEAVED) {
    __builtin_prefetch(&mi[idx], 0, 0);  // global_prefetch_b8: warm own line
  } else {
    __builtin_prefetch(&mp[idx], 0, 0);
  }

  const float x0 = (float)x;
  const float y0 = (float)y;

  float cx = x0, cy = y0;       // dc        (integration coordinate)
  float acx = x0, acy = y0;     // dc_acc    (starts at coord, per reference)
  bool invalid = false;

  const int n = n_steps_ptr[0]; // uniform scalar load (destination_frame)

  for (int s = 0; s < n; ++s) {
    float2 k1, k2, k3, k4;
    if (INTERLEAVED) {
      k1 = samp_interleaved(mi, cx, cy);
      k2 = samp_interleaved(mi, cx + 0.5f * k1.x, cy + 0.5f * k1.y);
      k3 = samp_interleaved(mi, cx + 0.5f * k2.x, cy + 0.5f * k2.y);
      k4 = samp_interleaved(mi, cx + k3.x, cy + k3.y);
    } else {
      k1 = samp_planar(mp, cx, cy);
      k2 = samp_planar(mp, cx + 0.5f * k1.x, cy + 0.5f * k1.y);
      k3 = samp_planar(mp, cx + 0.5f * k2.x, cy + 0.5f * k2.y);
      k4 = samp_planar(mp, cx + k3.x, cy + k3.y);
    }

    // Reference order: ((k1 + 2*k2) + 2*k3) + k4, then IEEE divide by 6.
    // (x2 products are exact, so fp-contract cannot change the result.)
    const float dx = (k1.x + 2.0f * k2.x + 2.0f * k3.x + k4.x) / 6.0f;
    const float dy = (k1.y + 2.0f * k2.y + 2.0f * k3.y + k4.y) / 6.0f;

    cx += dx;
    cy += dy;

    // oob on unclamped dc, sticky invalid, reset dc (but NOT dc_acc).
    const bool oob = (cx > (float)(kW - 1)) | (cx < 0.0f) |
                     (cy > (float)(kH - 1)) | (cy < 0.0f);
    invalid |= oob;
    if (invalid) { cx = x0; cy = y0; }

    acx += dx;   // dc_acc accumulates unconditionally (reference semantics)
    acy += dy;
  }

  // displacement = dc_acc - coord ; output layout (1,2,H,W) channel-planar.
  // NT stores: written once, never re-read -> keep L2 for the motion field.
  __builtin_nontemporal_store(acx - x0, &out[idx]);
  __builtin_nontemporal_store(acy - y0, &out[kHW + idx]);
}

extern "C" void kernel_launch(void* const* d_in, const int* in_sizes, int n_in,
                              void* d_out, int out_size, void* d_ws, size_t ws_size,
                              hipStream_t stream) {
  const float* motion   = (const float*)d_in[0];  // (1,2,H,W) f32
  const int*   n_frames = (const int*)d_in[1];    // scalar int on device
  float*       out      = (float*)d_out;          // (1,2,H,W) f32

  const dim3 grid(kW / 256, kH);   // 15 x 2160 blocks, 256 threads = 8 waves
  const dim3 block(256);

  if (ws_size >= (size_t)kHW * sizeof(float2)) {
    float2* mi = (float2*)d_ws;
    rk4_interleave_kernel<<<kHW / 256, 256, 0, stream>>>(motion, mi);
    rk4_integrate_kernel<true><<<grid, block, 0, stream>>>(mi, n_frames, out);
  } else {
    rk4_integrate_kernel<false><<<grid, block, 0, stream>>>(motion, n_frames, out);
  }
}